// SpatialTemporalGCN_65592740544775
// MI455X (gfx1250) — compile-verified
//
#include <hip/hip_runtime.h>

typedef float v2f __attribute__((ext_vector_type(2)));
typedef float v8f __attribute__((ext_vector_type(8)));

#define NNODES 20000
#define NEDGES 160000
#define MCH    8
#define FIN_C  32
#define HID_C  64
#define FOUT_C 32

static __device__ inline v8f wmma4(v2f a, v2f b, v8f c) {
#if __has_builtin(__builtin_amdgcn_wmma_f32_16x16x4_f32)
  // D = A(16x4,f32) x B(4x16,f32) + C(16x16,f32)
  return __builtin_amdgcn_wmma_f32_16x16x4_f32(false, a, false, b, (short)0, c, false, false);
#else
  // fallback so compilation never breaks; histogram (wmma=0) will flag this path
  c[0] += a[0] * b[0] + a[1] * b[1];
  return c;
#endif
}

static __device__ inline void atomic_add_f32(float* p, float v) {
  unsafeAtomicAdd(p, v);   // lowers to global_atomic_add_f32
}

// ---------------- degree / dinv -----------------------------------------

__global__ void k_init_deg(float* __restrict__ deg) {
  int i = blockIdx.x * 256 + threadIdx.x;
  if (i < NNODES) deg[i] = 1.0f;  // self-loop weight
}

__global__ void k_deg(const int* __restrict__ dst, float* __restrict__ deg) {
  int e = blockIdx.x * 256 + threadIdx.x;
  if (e < NEDGES) atomic_add_f32(&deg[dst[e]], 1.0f);
}

__global__ void k_dinv(const float* __restrict__ deg, float* __restrict__ dinv) {
  int i = blockIdx.x * 256 + threadIdx.x;
  if (i < NNODES) {
    float d = deg[i];
    dinv[i] = (d > 0.0f) ? rsqrtf(d) : 0.0f;
  }
}

// ---------------- WMMA GEMM: hs = (A @ W) * dinv[node] ------------------
// A rows are (node n, channel m) with r = n*MCH + m.
// Layer 0 additionally computes res = A @ Wr + br (unscaled residual) and
// reads A from the untransposed input x[m][n][f].

template <int FINT, int FOUTT, bool L0>
__global__ __launch_bounds__(256) void gcn_gemm(
    const float* __restrict__ A, const float* __restrict__ W,
    const float* __restrict__ Wr, const float* __restrict__ br,
    const float* __restrict__ dinv, float* __restrict__ hs,
    float* __restrict__ res) {
  constexpr int NT = FOUTT / 16;
  const int lane = threadIdx.x & 31;
  const int wave = threadIdx.x >> 5;
  const int rt   = (blockIdx.x * 8 + wave) * 16;   // row-tile base
  const int mrow = lane & 15;                      // A row within tile / B,C column
  const int half = lane >> 4;                      // K-half selector

  const float* arow;
  if constexpr (L0) {
    int r  = rt + mrow;
    int n  = r >> 3;   // r / MCH
    int mm = r & 7;    // r % MCH
    arow = A + ((size_t)mm * NNODES + n) * FINT;   // on-the-fly transpose of x
  } else {
    arow = A + (size_t)(rt + mrow) * FINT;
  }

  v8f acc[NT]  = {};
  v8f accr[NT] = {};

  #pragma unroll
  for (int k0 = 0; k0 < FINT; k0 += 4) {
    const int k = k0 + 2 * half;                   // A/B ISA layout: K = v + 2*half
    v2f a = *(const v2f*)(arow + k);
    #pragma unroll
    for (int t = 0; t < NT; ++t) {
      const int col = t * 16 + mrow;
      v2f b = { W[(size_t)k * FOUTT + col], W[(size_t)(k + 1) * FOUTT + col] };
      acc[t] = wmma4(a, b, acc[t]);
      if constexpr (L0) {
        v2f b2 = { Wr[(size_t)k * FOUTT + col], Wr[(size_t)(k + 1) * FOUTT + col] };
        accr[t] = wmma4(a, b2, accr[t]);
      }
    }
  }

  // C/D layout: VGPR j, lane l -> row = j + 8*(l/16), col = l%16
  #pragma unroll
  for (int t = 0; t < NT; ++t) {
    const int col = t * 16 + mrow;
    #pragma unroll
    for (int j = 0; j < 8; ++j) {
      int   rr = rt + j + 8 * half;
      float dv = dinv[rr >> 3];                     // node of row rr
      hs[(size_t)rr * FOUTT + col] = acc[t][j] * dv;
      if constexpr (L0) res[(size_t)rr * FOUTT + col] = accr[t][j] + br[col];
    }
  }
}

// ---------------- edge scatter: agg[dst] += hs[src] ---------------------

template <int BF>
__global__ __launch_bounds__(256) void k_scatter(
    const int* __restrict__ src, const int* __restrict__ dst,
    const float* __restrict__ hs, float* __restrict__ agg) {
  size_t t = (size_t)blockIdx.x * 256 + threadIdx.x;
  if (t >= (size_t)NEDGES * BF) return;
  int e = (int)(t / BF);
  int j = (int)(t % BF);
  int s = src[e];
  int d = dst[e];
  atomic_add_f32(&agg[(size_t)d * BF + j], hs[(size_t)s * BF + j]);
}

// ---------------- finalize: relu(agg*dinv + b) [+ res] ------------------

template <int F>
__global__ __launch_bounds__(256) void k_finalize(
    const float* __restrict__ agg, const float* __restrict__ dinv,
    const float* __restrict__ bias, const float* __restrict__ res,
    float* __restrict__ out, int total) {
  int i = blockIdx.x * 256 + threadIdx.x;
  if (i >= total) return;
  int   f = i % F;
  int   n = i / (F * MCH);
  float v = fmaxf(agg[i] * dinv[n] + bias[f], 0.0f);
  if (res) v += res[i];
  out[i] = v;
}

__global__ __launch_bounds__(256) void k_finalize_out(
    const float* __restrict__ agg, const float* __restrict__ dinv,
    const float* __restrict__ bias, float* __restrict__ out) {
  int i = blockIdx.x * 256 + threadIdx.x;
  if (i >= NNODES * MCH * FOUT_C) return;
  int   f = i % FOUT_C;
  int   m = (i / FOUT_C) % MCH;
  int   n = i / (FOUT_C * MCH);
  float v = fmaxf(agg[i] * dinv[n] + bias[f], 0.0f);
  out[((size_t)m * NNODES + n) * FOUT_C + f] = v;   // [B,S,N,FOUT]
}

// ---------------- host-side launch --------------------------------------

extern "C" void kernel_launch(void* const* d_in, const int* in_sizes, int n_in,
                              void* d_out, int out_size, void* d_ws, size_t ws_size,
                              hipStream_t stream) {
  const float* x  = (const float*)d_in[0];
  const int*   ei = (const int*)d_in[1];
  const float* W0 = (const float*)d_in[2];
  const float* b0 = (const float*)d_in[3];
  const float* W1 = (const float*)d_in[4];
  const float* b1 = (const float*)d_in[5];
  const float* W2 = (const float*)d_in[6];
  const float* b2 = (const float*)d_in[7];
  const float* Wr = (const float*)d_in[8];
  const float* br = (const float*)d_in[9];
  const int* src = ei;
  const int* dst = ei + NEDGES;

  const size_t NMH = (size_t)NNODES * MCH * HID_C;   // 10,240,000
  float* ws   = (float*)d_ws;
  float* deg  = ws;
  float* dinv = deg + NNODES;
  float* hs   = dinv + NNODES;
  float* agg  = hs + NMH;
  float* res  = agg + NMH;
  float* h    = res + NMH;

  // gcn_norm
  k_init_deg<<<(NNODES + 255) / 256, 256, 0, stream>>>(deg);
  k_deg<<<(NEDGES + 255) / 256, 256, 0, stream>>>(dst, deg);
  k_dinv<<<(NNODES + 255) / 256, 256, 0, stream>>>(deg, dinv);

  const int gemmBlocks = (NNODES * MCH) / (16 * 8);  // 1250
  const int totH = (int)NMH;

  // layer 0 (+ residual)
  gcn_gemm<FIN_C, HID_C, true><<<gemmBlocks, 256, 0, stream>>>(x, W0, Wr, br, dinv, hs, res);
  hipMemcpyAsync(agg, hs, NMH * sizeof(float), hipMemcpyDeviceToDevice, stream); // self-loop term
  k_scatter<MCH * HID_C><<<(unsigned)((size_t)NEDGES * MCH * HID_C / 256), 256, 0, stream>>>(src, dst, hs, agg);
  k_finalize<HID_C><<<(totH + 255) / 256, 256, 0, stream>>>(agg, dinv, b0, res, h, totH);

  // layer 1
  gcn_gemm<HID_C, HID_C, false><<<gemmBlocks, 256, 0, stream>>>(h, W1, nullptr, nullptr, dinv, hs, nullptr);
  hipMemcpyAsync(agg, hs, NMH * sizeof(float), hipMemcpyDeviceToDevice, stream);
  k_scatter<MCH * HID_C><<<(unsigned)((size_t)NEDGES * MCH * HID_C / 256), 256, 0, stream>>>(src, dst, hs, agg);
  k_finalize<HID_C><<<(totH + 255) / 256, 256, 0, stream>>>(agg, dinv, b1, nullptr, h, totH);

  // layer 2 -> output
  const size_t NMF = (size_t)NNODES * MCH * FOUT_C;  // 5,120,000
  gcn_gemm<HID_C, FOUT_C, false><<<gemmBlocks, 256, 0, stream>>>(h, W2, nullptr, nullptr, dinv, hs, nullptr);
  hipMemcpyAsync(agg, hs, NMF * sizeof(float), hipMemcpyDeviceToDevice, stream);
  k_scatter<MCH * FOUT_C><<<(unsigned)((size_t)NEDGES * MCH * FOUT_C / 256), 256, 0, stream>>>(src, dst, hs, agg);
  k_finalize_out<<<(unsigned)((NMF + 255) / 256), 256, 0, stream>>>(agg, dinv, b2, (float*)d_out);
}